// DroppedLinearV2_74938589381030
// MI455X (gfx1250) — compile-verified
//
#include <hip/hip_runtime.h>

typedef float v2f __attribute__((ext_vector_type(2)));
typedef float v8f __attribute__((ext_vector_type(8)));

#define DIM    2048   // IN4 == OUT4
#define BLK    512    // diagonal block size (DIM/4)
#define MT     128    // rows per workgroup tile
#define NT     128    // output cols per workgroup tile
#define KT     32     // K slice per LDS stage
#define LDSS   36     // padded LDS row stride in dwords (16B-aligned, conflict-free)

__global__ __launch_bounds__(256)
void blockdiag_linear_wmma_f32(const float* __restrict__ x,
                               const float* __restrict__ W,
                               const float* __restrict__ bias,
                               float* __restrict__ out) {
  __shared__ float Xs[MT * LDSS];
  __shared__ float Ws[NT * LDSS];

  const int tid  = threadIdx.x;
  const int wave = tid >> 5;
  const int lane = tid & 31;
  const int l16  = lane & 15;
  const int lhi  = lane >> 4;          // 0 or 1

  const int row0    = blockIdx.x * MT;             // x-row base
  const int blk     = blockIdx.z;                  // diagonal block id
  const int colBase = blk * BLK + blockIdx.y * NT; // output column base
  const int kBase   = blk * BLK;                   // k base for this block

  const int wr = wave >> 1;            // 0..3 : 32-row group within tile
  const int wc = wave & 1;             // 0..1 : 64-col group within tile

  v8f acc[2][4];
#pragma unroll
  for (int mt = 0; mt < 2; ++mt)
#pragma unroll
    for (int nt = 0; nt < 4; ++nt)
      acc[mt][nt] = (v8f){0.f, 0.f, 0.f, 0.f, 0.f, 0.f, 0.f, 0.f};

  // Cooperative loader mapping: 256 threads, each loads 4x float4 per matrix per stage.
  const int lr = tid >> 3;             // 0..31 : row within a 32-row pass
  const int lq = (tid & 7) * 4;        // dword column 0,4,...,28

  const float* xg = x + (size_t)row0 * DIM + kBase;
  const float* wg = W + (size_t)colBase * DIM + kBase;

  float4 xreg[4], wreg[4];

  // Prefetch stage 0.
#pragma unroll
  for (int p = 0; p < 4; ++p) {
    const int r = lr + p * 32;
    xreg[p] = *(const float4*)(xg + (size_t)r * DIM + lq);
    wreg[p] = *(const float4*)(wg + (size_t)r * DIM + lq);
  }

  const int nStages = BLK / KT;        // 16
  for (int s = 0; s < nStages; ++s) {
    __syncthreads();                   // previous stage's compute done reading LDS
#pragma unroll
    for (int p = 0; p < 4; ++p) {
      const int r = lr + p * 32;
      *(float4*)(&Xs[r * LDSS + lq]) = xreg[p];
      *(float4*)(&Ws[r * LDSS + lq]) = wreg[p];
    }
    __syncthreads();                   // LDS tile visible to all waves

    if (s + 1 < nStages) {             // issue next-stage global loads early
      const float* xn = xg + (s + 1) * KT;
      const float* wn = wg + (s + 1) * KT;
#pragma unroll
      for (int p = 0; p < 4; ++p) {
        const int r = lr + p * 32;
        xreg[p] = *(const float4*)(xn + (size_t)r * DIM + lq);
        wreg[p] = *(const float4*)(wn + (size_t)r * DIM + lq);
      }
    }

    // Compute: 8 micro-steps of K=4, 8 WMMAs each.
#pragma unroll
    for (int k0 = 0; k0 < KT; k0 += 4) {
      const int kidx = k0 + 2 * lhi;   // lanes 0-15: K=k0,k0+1; lanes 16-31: K=k0+2,k0+3
      v2f afrag[2], bfrag[4];
#pragma unroll
      for (int mt = 0; mt < 2; ++mt) {
        const int rrow = wr * 32 + mt * 16 + l16;     // A: M = l16 within tile
        afrag[mt] = *(const v2f*)(&Xs[rrow * LDSS + kidx]);
      }
#pragma unroll
      for (int nt = 0; nt < 4; ++nt) {
        const int crow = wc * 64 + nt * 16 + l16;     // B: N = l16 within tile
        bfrag[nt] = *(const v2f*)(&Ws[crow * LDSS + kidx]);
      }
#pragma unroll
      for (int mt = 0; mt < 2; ++mt)
#pragma unroll
        for (int nt = 0; nt < 4; ++nt)
          acc[mt][nt] = __builtin_amdgcn_wmma_f32_16x16x4_f32(
              false, afrag[mt], false, bfrag[nt],
              (short)0, acc[mt][nt], false, false);
    }
  }

  // Epilogue: bias + store. C/D layout: lane holds column N = l16 (of a 16x16 tile),
  // rows = vgpr + 8*lhi.
#pragma unroll
  for (int mt = 0; mt < 2; ++mt) {
#pragma unroll
    for (int nt = 0; nt < 4; ++nt) {
      const int col = colBase + wc * 64 + nt * 16 + l16;
      const float bv = bias[col];
      const int rbase = row0 + wr * 32 + mt * 16 + 8 * lhi;
      float* op = out + (size_t)rbase * DIM + col;
      v8f a = acc[mt][nt];
#pragma unroll
      for (int vr = 0; vr < 8; ++vr)
        op[(size_t)vr * DIM] = a[vr] + bv;
    }
  }
}

extern "C" void kernel_launch(void* const* d_in, const int* in_sizes, int n_in,
                              void* d_out, int out_size, void* d_ws, size_t ws_size,
                              hipStream_t stream) {
  const float* x = (const float*)d_in[0];   // (N, 2048) f32
  const float* W = (const float*)d_in[1];   // (2048, 2048) f32
  const float* b = (const float*)d_in[2];   // (2048,) f32
  float* out = (float*)d_out;               // (N, 2048) f32

  const int nrows = in_sizes[0] / DIM;      // 8192
  dim3 grid(nrows / MT, BLK / NT, 4);       // (64, 4, 4)
  dim3 block(256, 1, 1);
  hipLaunchKernelGGL(blockdiag_linear_wmma_f32, grid, block, 0, stream,
                     x, W, b, out);
}